// SoftWindow_33646773797680
// MI455X (gfx1250) — compile-verified
//
#include <hip/hip_runtime.h>
#include <math.h>

// Problem constants (match reference)
#define Bn 256
#define Hn 1024
#define Un 1024
#define Cn 512
#define Kn 10
#define NP 30   // 3*K output params per batch row

typedef __attribute__((ext_vector_type(2))) float v2f;
typedef __attribute__((ext_vector_type(4))) float f32x4;
typedef __attribute__((ext_vector_type(8))) float v8f;

__device__ __forceinline__ float softplus_f(float x) {
  // stable softplus: exp underflows to 0 for very negative x -> log1p(0)=0
  return (x > 20.0f) ? x : log1pf(__expf(x));
}

// ---------------------------------------------------------------------------
// Kernel A: params = h @ W.T + b via V_WMMA_F32_16X16X4_F32, then softplus
// epilogue producing alpha/beta/kappa. M=256 (16 tiles) x N=32 (2 tiles,
// cols 30/31 are padding) -> 32 waves, one 16x16 tile each, K=1024 step 4.
// A layout (f32 16x4): lane L holds row M=L%16; VGPR0/1 = K = (L/16)*2 +{0,1}
// B layout (f32 4x16): lane L holds col N=L%16;  VGPR0/1 = K = (L/16)*2 +{0,1}
// D layout (f32 16x16): elem v at lane L -> row = v + (L/16)*8, col = L%16
// Padded columns (30,31) read clamped row 29 (finite garbage); the epilogue
// guard j < NP makes that harmless, so no per-iteration mask is needed.
// ---------------------------------------------------------------------------
__global__ __launch_bounds__(32) void params_wmma_kernel(
    const float* __restrict__ h, const float* __restrict__ k_prev,
    const float* __restrict__ W, const float* __restrict__ bias,
    float* __restrict__ ws_alpha, float* __restrict__ ws_beta,
    float* __restrict__ ws_kappa, float* __restrict__ out_kappa)
{
  const int tile = blockIdx.x;        // 0..31
  const int m0 = (tile >> 1) * 16;    // batch-row tile base
  const int n0 = (tile & 1) * 16;     // param-col tile base
  const int lane = threadIdx.x;       // 0..31, full wave (EXEC all ones)
  const int lm   = lane & 15;
  const int kq   = (lane >> 4) * 2;   // K sub-offset: 0 or 2

  const float* hrow = h + (size_t)(m0 + lm) * Hn;
  const int   wcol  = n0 + lm;
  const int   wc    = (wcol < NP) ? wcol : (NP - 1);   // clamp (branchless)
  const float* wrow = W + (size_t)wc * Hn;

  v8f acc = {};
  for (int k0 = 0; k0 < Hn; k0 += 4) {
    v2f a  = *(const v2f*)(hrow + k0 + kq);   // 8B aligned (kq even)
    v2f bv = *(const v2f*)(wrow + k0 + kq);
    acc = __builtin_amdgcn_wmma_f32_16x16x4_f32(
        /*neg_a=*/false, a, /*neg_b=*/false, bv,
        /*c_mod=*/(short)0, acc, /*reuse_a=*/false, /*reuse_b=*/false);
  }

  // Epilogue: bias + softplus transforms, scatter alpha/beta/kappa
  const int j = n0 + lm;
  #pragma unroll
  for (int v = 0; v < 8; ++v) {
    const int row = m0 + v + ((lane >> 4) << 3);
    if (j < NP) {
      const float val = acc[v] + bias[j];
      const float sp  = softplus_f(val);
      if (j < Kn) {
        ws_alpha[row * Kn + j] = sp + 1e-4f;
      } else if (j < 2 * Kn) {
        ws_beta[row * Kn + (j - Kn)] = fminf(fmaxf(sp, 0.1f), 10.0f);
      } else {
        const int kk = j - 2 * Kn;
        const float kp = k_prev[row * Kn + kk] + sp * 0.1f;
        ws_kappa[row * Kn + kk]  = kp;
        out_kappa[row * Kn + kk] = kp;   // third tuple output
      }
    }
  }
}

// ---------------------------------------------------------------------------
// Kernel B: fused phi + window partials. Grid (Bn, S): block (b, s) owns the
// u-range [s*Un/S, (s+1)*Un/S). 512 threads = 16 waves. Phase 1 computes the
// phi slice into LDS (and d_out). Phase 2 streams its slice of c_seq[b,:,:]
// with coalesced non-temporal b128 loads: group g=tid/128 owns u rows
// ubeg+g, ubeg+g+4, ...; lanes tid%128 own 4 consecutive C columns each.
// Phase 3: deterministic LDS reduction of the 4 u-groups, then write either
// the final window row (S==1) or a partial row (S>1, combined by kernel C).
// ---------------------------------------------------------------------------
__global__ __launch_bounds__(512) void window_kernel(
    const float* __restrict__ c_seq,
    const float* __restrict__ ws_alpha, const float* __restrict__ ws_beta,
    const float* __restrict__ ws_kappa,
    float* __restrict__ dst,          // S==1: out_window ; S>1: partials [B,S,C]
    float* __restrict__ out_phi,
    int S)
{
  __shared__ float sA[Kn], sB[Kn], sK[Kn];
  __shared__ float sphi[Un];
  __shared__ f32x4 sred[512];

  const int b    = blockIdx.x;
  const int s    = blockIdx.y;
  const int tid  = threadIdx.x;
  const int ulen = Un / S;
  const int ubeg = s * ulen;

  if (tid < Kn) {
    sA[tid] = ws_alpha[b * Kn + tid];
    sB[tid] = ws_beta [b * Kn + tid];
    sK[tid] = ws_kappa[b * Kn + tid];
  }
  __syncthreads();

  // Phase 1: phi[b,u] = sum_k alpha_k * exp(-beta_k * (kappa_k - u)^2)
  for (int du = tid; du < ulen; du += 512) {
    const int u = ubeg + du;
    const float uu = (float)u;
    float p = 0.0f;
    #pragma unroll
    for (int k = 0; k < Kn; ++k) {
      const float d = sK[k] - uu;
      p += sA[k] * __expf(-sB[k] * d * d);
    }
    sphi[du] = p;
    out_phi[(size_t)b * Un + u] = p;   // second tuple output
  }
  __syncthreads();

  // Phase 2: stream c_seq row-slice, accumulate phi-weighted columns
  const int g  = tid >> 7;     // 0..3: u-row within each group of 4
  const int cg = tid & 127;    // column group: cols [4*cg, 4*cg+3]
  const f32x4* base =
      (const f32x4*)(c_seq + ((size_t)b * Un + ubeg) * Cn) + cg;

  f32x4 acc = {};
  #pragma unroll 4
  for (int du = g; du < ulen; du += 4) {
    const f32x4 v = __builtin_nontemporal_load(base + (size_t)du * (Cn / 4));
    acc += sphi[du] * v;
  }

  // Phase 3: fixed-order reduction across the 4 u-groups (deterministic)
  sred[tid] = acc;
  __syncthreads();
  if (tid < 128) {
    const f32x4 r = sred[tid] + sred[tid + 128] + sred[tid + 256] + sred[tid + 384];
    f32x4* drow = (f32x4*)(dst + ((size_t)b * S + s) * Cn);
    drow[tid] = r;
  }
}

// ---------------------------------------------------------------------------
// Kernel C: combine S u-split partials into the final window row.
// Fixed summation order -> deterministic.
// ---------------------------------------------------------------------------
__global__ __launch_bounds__(128) void combine_kernel(
    const float* __restrict__ partials,   // [B,S,C]
    float* __restrict__ out_window,       // [B,C]
    int S)
{
  const int b   = blockIdx.x;
  const int tid = threadIdx.x;            // 128 threads x f32x4 = 512 cols
  const f32x4* p = (const f32x4*)(partials + (size_t)b * S * Cn) + tid;
  f32x4 r = p[0];
  for (int s = 1; s < S; ++s) r += p[(size_t)s * (Cn / 4)];
  *((f32x4*)(out_window + (size_t)b * Cn) + tid) = r;
}

// ---------------------------------------------------------------------------
extern "C" void kernel_launch(void* const* d_in, const int* in_sizes, int n_in,
                              void* d_out, int out_size, void* d_ws, size_t ws_size,
                              hipStream_t stream) {
  (void)in_sizes; (void)n_in; (void)out_size;

  const float* h      = (const float*)d_in[0];   // [B,H]
  const float* c_seq  = (const float*)d_in[1];   // [B,U,C]
  const float* k_prev = (const float*)d_in[2];   // [B,K]
  const float* W      = (const float*)d_in[3];   // [3K,H]
  const float* bias   = (const float*)d_in[4];   // [3K]

  float* out        = (float*)d_out;
  float* out_window = out;                        // B*C
  float* out_phi    = out + (size_t)Bn * Cn;      // B*U
  float* out_kappa  = out + (size_t)Bn * Cn + (size_t)Bn * Un;  // B*K

  float* ws       = (float*)d_ws;
  float* ws_alpha = ws;                           // B*K
  float* ws_beta  = ws + Bn * Kn;                 // B*K
  float* ws_kappa = ws + 2 * Bn * Kn;             // B*K
  float* ws_part  = ws + 3 * Bn * Kn;             // [B,S,C] partials (16B-aligned)

  params_wmma_kernel<<<32, 32, 0, stream>>>(h, k_prev, W, bias,
                                            ws_alpha, ws_beta, ws_kappa,
                                            out_kappa);

  // U-split factor: 4x more blocks for machine fill; fall back to single-pass
  // if the scratch buffer cannot hold the partials.
  const int S = 4;
  const size_t need = (size_t)(3 * Bn * Kn) * sizeof(float)
                    + (size_t)Bn * S * Cn * sizeof(float);
  if (ws_size >= need) {
    window_kernel<<<dim3(Bn, S), 512, 0, stream>>>(
        c_seq, ws_alpha, ws_beta, ws_kappa, ws_part, out_phi, S);
    combine_kernel<<<Bn, 128, 0, stream>>>(ws_part, out_window, S);
  } else {
    window_kernel<<<dim3(Bn, 1), 512, 0, stream>>>(
        c_seq, ws_alpha, ws_beta, ws_kappa, out_window, out_phi, 1);
  }
}